// CombinedRotaryEmbedding_712964571680
// MI455X (gfx1250) — compile-verified
//
#include <hip/hip_runtime.h>
#include <hip/hip_bf16.h>
#include <math.h>

typedef __attribute__((ext_vector_type(2))) float v2f;
typedef __attribute__((ext_vector_type(8))) float v8f;

// Problem constants (fixed by the reference)
#define HD      128            // head dim
#define NROT    64
#define NROWS   262144         // 4 * 2048 * 32
#define NTILES  (NROWS / 16)   // 16384
#define TPW     8              // row-tiles per workgroup
#define NWG     (NTILES / TPW) // 2048

// ---------------------------------------------------------------------------
// Setup kernel: fold the 64 sequential Givens rotations and the dense
// rotation_matrix into one 128x128 matrix, stored TRANSPOSED in d_ws:
//   MT[n][k] = M[k][n],  M = (R_1 R_2 ... R_64) @ rotation_matrix
// One block, 128 threads. Thread r owns row r of G -> no barriers needed
// during the rotation phase (rotations only mix columns within a row).
// ---------------------------------------------------------------------------
__global__ __launch_bounds__(128) void build_mt_kernel(
    const float* __restrict__ thetas,
    const float* __restrict__ pairs,        // (64,2) float, cast->int
    const float* __restrict__ theta_scale,  // (1,)
    const float* __restrict__ rotmat,       // (128,128)
    float* __restrict__ mt)                 // d_ws: (128,128) = M^T
{
    __shared__ float G[HD * HD];            // 64 KB
    const int r = threadIdx.x;              // 0..127

    for (int c = 0; c < HD; ++c)
        G[r * HD + c] = (r == c) ? 1.0f : 0.0f;

    const float ts = theta_scale[0];
    for (int k = 0; k < NROT; ++k) {
        const int   i  = (int)pairs[2 * k + 0];   // truncation matches astype(int32)
        const int   j  = (int)pairs[2 * k + 1];
        const float th = thetas[k] * ts;
        const float ct = cosf(th);
        const float st = sinf(th);
        const float gi = G[r * HD + i];
        const float gj = G[r * HD + j];
        if (i == j) {
            G[r * HD + i] = gi * ct;              // ref sets col i then j; same col
        } else {
            G[r * HD + i] =  gi * ct + gj * st;
            G[r * HD + j] = -gi * st + gj * ct;
        }
    }
    __syncthreads();

    // MT[n][k] = sum_c G[k][c] * rotmat[c][n]; thread n computes MT row n.
    // rotmat[c*128+n] reads are coalesced across threads for each fixed c.
    const int n = threadIdx.x;
    for (int k = 0; k < HD; ++k) {
        float acc = 0.0f;
        for (int c = 0; c < HD; ++c)
            acc = fmaf(G[k * HD + c], rotmat[c * HD + n], acc);
        mt[n * HD + k] = acc;
    }
}

// ---------------------------------------------------------------------------
// Main kernel: out = RoPE( X @ M ), X = (262144, 128) f32, streamed.
// 256 threads = 8 waves; each wave owns a 16-column slice of M (B fragments
// held in 64 VGPRs for the whole kernel). Each workgroup processes TPW
// 16-row tiles: x-tile -> LDS (coalesced b128), 32 x v_wmma_f32_16x16x4_f32,
// accumulators -> LDS y-tile, RoPE epilogue with coalesced 512B stores.
// A 16-row tile never straddles a sequence position (32 rows per (b,s)),
// so sin/cos are constant per tile per channel.
// ---------------------------------------------------------------------------
__global__ __launch_bounds__(256) void rotary_main_kernel(
    const float* __restrict__ x,
    const float* __restrict__ inv_freq,     // (64,)
    const float* __restrict__ mt,           // (128,128) = M^T from d_ws
    float* __restrict__ out)
{
    __shared__ float xt[16 * HD];           // 8 KB: x tile
    __shared__ float yt[16 * HD];           // 8 KB: y = x@M tile

    const int tid   = threadIdx.x;
    const int lane  = tid & 31;
    const int wave  = tid >> 5;             // 0..7
    const int n0    = wave << 4;            // column block base
    const int lhalf = lane >> 4;            // 0 or 1
    const int l15   = lane & 15;

    // --- Preload B fragments (this wave's 128x16 slice of M) -------------
    // ISA layout (f32 16x16x4): lanes 0-15 hold K-pair {0,1}, lanes 16-31
    // hold {2,3}; VGPR v selects K within the pair. With MT stored
    // transposed, each fragment is one contiguous float2 per lane.
    v2f bfrag[32];
    {
        const float* mrow = mt + (n0 + l15) * HD + (lhalf << 1);
#pragma unroll
        for (int kk = 0; kk < 32; ++kk)
            bfrag[kk] = *(const v2f*)(mrow + kk * 4);
    }

    // --- Per-thread RoPE parameters (epilogue mapping) --------------------
    const int   c    = tid & 127;           // output channel
    const int   rb   = tid >> 7;            // 0 or 1 (row parity base)
    const int   kidx = c & 63;              // rotation-pair index
    const float invf = inv_freq[kidx];

    for (int it = 0; it < TPW; ++it) {
        const int  tile = blockIdx.x * TPW + it;
        const long r0   = (long)tile * 16;  // first row of this tile

        // --- Stage 16x128 x-tile into LDS (coalesced b128 loads) ---------
        {
            const float4* xg = (const float4*)(x + r0 * HD);
            float4*       xl = (float4*)xt;
            xl[tid]       = xg[tid];
            xl[tid + 256] = xg[tid + 256];
        }
        __syncthreads();

        // --- 16x16 tile of X@M via 32 chained f32 WMMAs ------------------
        v8f acc = {};
        {
            const float* arow = xt + l15 * HD + (lhalf << 1);
#pragma unroll
            for (int kk = 0; kk < 32; ++kk) {
                v2f a = *(const v2f*)(arow + kk * 4);
                acc = __builtin_amdgcn_wmma_f32_16x16x4_f32(
                    /*neg_a=*/false, a, /*neg_b=*/false, bfrag[kk],
                    /*c_mod=*/(short)0, acc,
                    /*reuse_a=*/false, /*reuse_b=*/false);
            }
        }

        // --- Spill accumulators to LDS y-tile ----------------------------
        // D layout: VGPR p, lanes 0-15 -> row p, lanes 16-31 -> row p+8.
#pragma unroll
        for (int p = 0; p < 8; ++p)
            yt[(p + (lhalf << 3)) * HD + n0 + l15] = acc[p];
        __syncthreads();

        // --- RoPE epilogue + coalesced stores ----------------------------
        // out[r][k]    = y[2k]*cos - y[2k+1]*sin   (k < 64)
        // out[r][64+k] = y[2k]*sin + y[2k+1]*cos
        {
            const int   spos = (int)((r0 >> 5) & 2047);
            const float ang  = (float)spos * invf;
            const float sv   = sinf(ang);
            const float cv   = cosf(ang);
            float* orow = out + r0 * HD;
#pragma unroll
            for (int p = 0; p < 8; ++p) {
                const int   row = rb + (p << 1);
                const float y1  = yt[row * HD + 2 * kidx];
                const float y2  = yt[row * HD + 2 * kidx + 1];
                const float v   = (c < 64) ? (y1 * cv - y2 * sv)
                                           : (y1 * sv + y2 * cv);
                orow[row * HD + c] = v;      // 128 consecutive lanes/chan
            }
        }
        // next iteration's first __syncthreads separates these yt reads
        // from the next tile's yt writes; xt is a disjoint buffer.
    }
}

extern "C" void kernel_launch(void* const* d_in, const int* in_sizes, int n_in,
                              void* d_out, int out_size, void* d_ws, size_t ws_size,
                              hipStream_t stream) {
    const float* x        = (const float*)d_in[0];
    const float* thetas   = (const float*)d_in[1];
    const float* pairs    = (const float*)d_in[2];
    const float* tscale   = (const float*)d_in[3];
    const float* rotmat   = (const float*)d_in[4];
    const float* inv_freq = (const float*)d_in[5];
    float*       mt       = (float*)d_ws;           // 128*128*4 = 64 KB
    float*       out      = (float*)d_out;

    build_mt_kernel<<<1, 128, 0, stream>>>(thetas, pairs, tscale, rotmat, mt);
    rotary_main_kernel<<<NWG, 256, 0, stream>>>(x, inv_freq, mt, out);
}